// DictionaryLearningTokenized_78365973283477
// MI455X (gfx1250) — compile-verified
//
#include <hip/hip_runtime.h>
#include <math.h>
#include <stdint.h>

typedef __attribute__((ext_vector_type(2))) float v2f;
typedef __attribute__((ext_vector_type(8))) float v8f;

#define N_EMBED   512
#define EMBED_DIM 64
#define K_SPARSE  4
#define N_BINS    33
#define NSIG      16384        // 16*32*32 signals
#define SIG_PER_WG 32
#define ZQ_ELEMS  1048576      // 16*64*32*32
#define LOSS_OFF  ZQ_ELEMS
#define TOK_OFF   (ZQ_ELEMS + 1)

#define HBAR_FLOATS (SIG_PER_WG * N_EMBED)          // 16384 floats = 64 KB
#define DT_FLOATS   (N_EMBED * EMBED_DIM)           // 32768 floats = 128 KB
#define LDS_BYTES   ((HBAR_FLOATS + DT_FLOATS) * 4) // 192 KB dynamic LDS

// ---------------------------------------------------------------------------
// Kernel 1: normalize dictionary columns -> atom-major Dt[512][64]; zero acc.
// dictionary layout: [EMBED_DIM, N_EMBED] row-major -> dict[c*512 + n]
// ---------------------------------------------------------------------------
__global__ void prep_kernel(const float* __restrict__ dict,
                            float* __restrict__ Dt,
                            float* __restrict__ acc) {
    int n = threadIdx.x;
    if (n == 0) *acc = 0.f;
    if (n < N_EMBED) {
        float s = 0.f;
        for (int c = 0; c < EMBED_DIM; ++c) {
            float v = dict[c * N_EMBED + n];
            s += v * v;
        }
        float inv = 1.f / fmaxf(sqrtf(s), 1e-10f);
        for (int c = 0; c < EMBED_DIM; ++c)
            Dt[n * EMBED_DIM + c] = dict[c * N_EMBED + n] * inv;
    }
}

// ---------------------------------------------------------------------------
// Kernel 2: G = Dt * Dt^T  (512x512, K=64) with V_WMMA_F32_16X16X4_F32.
// One wave per 16x16 output tile; 4 waves per block; grid (8, 32).
// ---------------------------------------------------------------------------
__global__ __launch_bounds__(128) void gram_kernel(const float* __restrict__ Dt,
                                                   float* __restrict__ G) {
    int wave = threadIdx.x >> 5;
    int lane = threadIdx.x & 31;
    int half = lane >> 4;
    int l    = lane & 15;
    int tm = blockIdx.x * 4 + wave;   // 0..31
    int tn = blockIdx.y;              // 0..31

    int arow = tm * 16 + l;
    int brow = tn * 16 + l;
    v8f c = {};
    for (int k0 = 0; k0 < EMBED_DIM; k0 += 4) {
        int ka = k0 + half * 2;
        v2f a, b;
        a.x = Dt[arow * EMBED_DIM + ka];
        a.y = Dt[arow * EMBED_DIM + ka + 1];
        b.x = Dt[brow * EMBED_DIM + ka];
        b.y = Dt[brow * EMBED_DIM + ka + 1];
        c = __builtin_amdgcn_wmma_f32_16x16x4_f32(false, a, false, b,
                                                  (short)0, c, false, false);
    }
    for (int r = 0; r < 8; ++r) {
        int row = tm * 16 + half * 8 + r;
        G[row * N_EMBED + tn * 16 + l] = c[r];
    }
}

// ---------------------------------------------------------------------------
// Kernel 3: fused h_bar (WMMA into LDS) + OMP + quantize + reconstruct + loss.
// 512 blocks x 256 threads (8 waves); 32 signals per block.
// Dynamic LDS: hbar[32][512] (64 KB) | DtL[512][64] (128 KB)
// Dt is staged into LDS with GLOBAL_LOAD_ASYNC_TO_LDS_B128 (ASYNCcnt path).
// ---------------------------------------------------------------------------
__global__ __launch_bounds__(256) void omp_fused_kernel(const float* __restrict__ z_e,
                                                        const float* __restrict__ Dt,
                                                        const float* __restrict__ G,
                                                        float* __restrict__ out,
                                                        float* __restrict__ acc) {
    extern __shared__ float lds[];
    float* hbar = lds;                  // [32 * 512]
    float* DtL  = lds + HBAR_FLOATS;    // [512 * 64]

    int wave = threadIdx.x >> 5;
    int lane = threadIdx.x & 31;
    int half = lane >> 4;
    int l    = lane & 15;
    int wgBase = blockIdx.x * SIG_PER_WG;

    // ---- Stage Dt (128 KB) into LDS via async global->LDS DMA ----
    {
        unsigned ldsDst0 = (unsigned)(uintptr_t)(&DtL[0]);   // LDS byte address
        unsigned long long gbase = (unsigned long long)Dt;
        for (int i = threadIdx.x; i < DT_FLOATS / 4; i += 256) {
            unsigned dst  = ldsDst0 + (unsigned)i * 16u;
            unsigned voff = (unsigned)i * 16u;
            asm volatile("global_load_async_to_lds_b128 %0, %1, %2"
                         :: "v"(dst), "v"(voff), "s"(gbase) : "memory");
        }
        asm volatile("s_wait_asynccnt 0" ::: "memory");
    }

    // L2 warm-up hint for the Gram matrix rows (global_prefetch_b8)
    __builtin_prefetch(G + (threadIdx.x << 10), 0, 0);

    __syncthreads();   // publish DtL to all waves

    // ---- Phase 1: h_bar[32 x 512] tile via fp32 WMMA ----
    // wave 0-3 -> signal rows 0-15, wave 4-7 -> rows 16-31; 8 atom-tiles each.
    int sTile = wave >> 2;
    int sLoc  = sTile * 16 + l;
    int sGlob = wgBase + sLoc;
    int bimg  = sGlob >> 10;           // batch index
    int hw    = sGlob & 1023;          // h*32+w
    long zbase = (long)bimg * 65536 + hw;   // z_e[b, c, h, w] = zbase + c*1024

    // hoist A fragments (signals): reused across all 8 atom-tiles
    v2f afrag[16];
    for (int kk = 0; kk < 16; ++kk) {
        int ka = kk * 4 + half * 2;
        afrag[kk].x = z_e[zbase + (long)ka * 1024];
        afrag[kk].y = z_e[zbase + (long)(ka + 1) * 1024];
    }

    for (int t = 0; t < 8; ++t) {
        int nTile = (wave & 3) + 4 * t;
        int atom  = nTile * 16 + l;
        v8f c = {};
        for (int kk = 0; kk < 16; ++kk) {
            int ka = kk * 4 + half * 2;
            v2f b;
            b.x = DtL[atom * EMBED_DIM + ka];
            b.y = DtL[atom * EMBED_DIM + ka + 1];
            c = __builtin_amdgcn_wmma_f32_16x16x4_f32(false, afrag[kk], false, b,
                                                      (short)0, c, false, false);
        }
        for (int r = 0; r < 8; ++r) {
            int row = sTile * 16 + half * 8 + r;
            hbar[row * N_EMBED + nTile * 16 + l] = c[r];
        }
    }
    __syncthreads();

    // ---- Phase 2: OMP; each wave handles 4 signals ----
    float lossAcc = 0.f;
    for (int q = 0; q < 4; ++q) {
        int sL = wave * 4 + q;
        int sG = wgBase + sL;

        // lane owns atoms a = j*32 + lane (coalesced G reads)
        float hreg[16];
        for (int j = 0; j < 16; ++j)
            hreg[j] = hbar[sL * N_EMBED + j * 32 + lane];

        unsigned selMask = 0;
        int   sel[K_SPARSE];
        float xx[K_SPARSE];
        float Ls[K_SPARSE][K_SPARSE];

        for (int k = 1; k <= K_SPARSE; ++k) {
            // masked abs-argmax over this lane's atoms
            float bv = -1.f; int bi = 0x7fffffff;
            for (int j = 0; j < 16; ++j) {
                float av = ((selMask >> j) & 1u) ? -1.f : fabsf(hreg[j]);
                int idx = j * 32 + lane;
                if (av > bv || (av == bv && idx < bi)) { bv = av; bi = idx; }
            }
            // wave32 reduction (first-max tiebreak)
            for (int off = 16; off; off >>= 1) {
                float ov = __shfl_xor(bv, off, 32);
                int   oi = __shfl_xor(bi, off, 32);
                if (ov > bv || (ov == bv && oi < bi)) { bv = ov; bi = oi; }
            }
            sel[k - 1] = bi;
            if ((bi & 31) == lane) selMask |= 1u << (bi >> 5);

            // solve k x k normal equations G_S x = hbar_S (uniform scalar work)
            float Gs[K_SPARSE][K_SPARSE], hb[K_SPARSE], yv[K_SPARSE];
            for (int i = 0; i < k; ++i) {
                hb[i] = hbar[sL * N_EMBED + sel[i]];
                for (int j = 0; j <= i; ++j)
                    Gs[i][j] = G[sel[i] * N_EMBED + sel[j]];
            }
            for (int i = 0; i < k; ++i)
                for (int j = 0; j <= i; ++j) {
                    float s = Gs[i][j];
                    for (int m = 0; m < j; ++m) s -= Ls[i][m] * Ls[j][m];
                    Ls[i][j] = (i == j) ? sqrtf(fmaxf(s, 1e-12f)) : s / Ls[j][j];
                }
            for (int i = 0; i < k; ++i) {
                float s = hb[i];
                for (int m = 0; m < i; ++m) s -= Ls[i][m] * yv[m];
                yv[i] = s / Ls[i][i];
            }
            for (int i = k - 1; i >= 0; --i) {
                float s = yv[i];
                for (int m = i + 1; m < k; ++m) s -= Ls[m][i] * xx[m];
                xx[i] = s / Ls[i][i];
            }

            // residual correlations: h = h_bar - sum_i x_i * G[sel_i, :]
            if (k < K_SPARSE) {
                for (int j = 0; j < 16; ++j) {
                    float s = hbar[sL * N_EMBED + j * 32 + lane];
                    for (int i = 0; i < k; ++i)
                        s -= xx[i] * G[sel[i] * N_EMBED + j * 32 + lane];
                    hreg[j] = s;
                }
            }
        }

        // quantize coefficients to 33 bin centers; tokens
        float cq[K_SPARSE]; int tok[K_SPARSE];
        for (int i = 0; i < K_SPARSE; ++i) {
            float cc = fminf(fmaxf(xx[i], -2.f), 2.f);
            int bin = (int)rintf((cc + 2.f) * 8.f);
            bin = min(max(bin, 0), N_BINS - 1);
            cq[i]  = bin * 0.125f - 2.f;
            tok[i] = sel[i] * N_BINS + bin;
        }

        // reconstruction + loss partial; z_q layout == z_e layout
        int bimg2 = sG >> 10, hw2 = sG & 1023;
        long base2 = (long)bimg2 * 65536 + hw2;
        for (int h = 0; h < 2; ++h) {
            int ch = lane + 32 * h;
            float v = 0.f;
            for (int i = 0; i < K_SPARSE; ++i)
                v += cq[i] * DtL[sel[i] * EMBED_DIM + ch];
            long idx = base2 + (long)ch * 1024;
            out[idx] = v;                       // z_q_ste == z_q numerically
            float d = v - z_e[idx];
            lossAcc += d * d;
        }
        if (lane < K_SPARSE)
            out[TOK_OFF + sG * 4 + lane] = (float)tok[lane];
    }

    for (int off = 16; off; off >>= 1)
        lossAcc += __shfl_xor(lossAcc, off, 32);
    if (lane == 0) atomicAdd(acc, lossAcc);
}

// ---------------------------------------------------------------------------
// Kernel 4: finalize loss = (1 + COMMIT) * mean(sq err)
// ---------------------------------------------------------------------------
__global__ void loss_kernel(const float* __restrict__ acc, float* __restrict__ out) {
    out[LOSS_OFF] = 1.25f * (*acc) / (float)ZQ_ELEMS;
}

// ---------------------------------------------------------------------------
extern "C" void kernel_launch(void* const* d_in, const int* in_sizes, int n_in,
                              void* d_out, int out_size, void* d_ws, size_t ws_size,
                              hipStream_t stream) {
    const float* z_e  = (const float*)d_in[0];
    const float* dict = (const float*)d_in[1];
    float* out = (float*)d_out;

    // workspace: Dt[512*64] | G[512*512] | acc[1]
    float* Dt  = (float*)d_ws;
    float* G   = Dt + N_EMBED * EMBED_DIM;
    float* acc = G + N_EMBED * N_EMBED;

    prep_kernel<<<1, 512, 0, stream>>>(dict, Dt, acc);
    gram_kernel<<<dim3(8, 32), 128, 0, stream>>>(Dt, G);
    omp_fused_kernel<<<NSIG / SIG_PER_WG, 256, LDS_BYTES, stream>>>(z_e, Dt, G, out, acc);
    loss_kernel<<<1, 1, 0, stream>>>(acc, out);
}